// Hybrid_QML_Engine_47167330844883
// MI455X (gfx1250) — compile-verified
//
#include <hip/hip_runtime.h>
#include <math.h>

typedef __attribute__((ext_vector_type(2))) float v2f;
typedef __attribute__((ext_vector_type(8))) float v8f;

#define NQ       12
#define NSTATES  4096          // 2^12
#define NLAYERS  4
#define LDIM     128
#define BM       2048          // 64 * 32 rows
#define FOUT     96

// ---------------------------------------------------------------------------
// Phase 1: H = Xflat @ W_in^T + b_in + 1e-6   (full fp32 via V_WMMA_F32_16X16X4_F32)
// Xflat[r][l] = x[r>>5][l][r&31]  (x is (64,128,32))
// One wave computes one 16x16 tile of H; K loop = 32 WMMA steps of K=4.
// A frag (16x4, ISA 7.12.2): lane L -> M=L%16, VGPR v holds K = 2*(L/16)+v.
// B frag (4x16) assumed mirrored: lane L -> N=L%16, VGPR v holds K = 2*(L/16)+v,
//   i.e. contiguous float2 from W_in[j0+N][k0 + 2*(L/16)].
// C/D (16x16): lane L -> N=L%16, VGPR v holds M = v + 8*(L/16).
// ---------------------------------------------------------------------------
__global__ __launch_bounds__(256) void qml_gemm_wmma(
    const float* __restrict__ x,       // (64,128,32)
    const float* __restrict__ W_in,    // (4096,128)
    const float* __restrict__ b_in,    // (4096,)
    float* __restrict__ H)             // (2048,4096)
{
    const int lane = threadIdx.x & 31;
    const int wave = threadIdx.x >> 5;
    const int tile = blockIdx.x * 8 + wave;        // 128*256 = 32768 tiles
    const int mt = tile >> 8;                      // 0..127
    const int nt = tile & 255;                     // 0..255
    const int m0 = mt * 16, j0 = nt * 16;

    const int lm = lane & 15;
    const int kh = (lane >> 4) * 2;

    // A: strided gather (implicit transpose of x)
    const int rA = m0 + lm;
    const float* xbase = x + (size_t)(rA >> 5) * (LDIM * 32) + (rA & 31);
    // B: contiguous along K within a W_in row
    const float* wbase = W_in + (size_t)(j0 + lm) * LDIM + kh;

    v8f c = {};
    for (int k0 = 0; k0 < LDIM; k0 += 4) {
        v2f a, b;
        a.x = xbase[(k0 + kh) * 32];
        a.y = xbase[(k0 + kh + 1) * 32];
        const float2 bw = *(const float2*)(wbase + k0);
        b.x = bw.x; b.y = bw.y;
        c = __builtin_amdgcn_wmma_f32_16x16x4_f32(false, a, false, b,
                                                  (short)0, c, false, false);
    }

    const float bias = b_in[j0 + lm] + 1e-6f;
    float* hb = H + j0 + lm;
    const int mrow0 = m0 + (lane >> 4) * 8;
#pragma unroll
    for (int v = 0; v < 8; ++v)
        hb[(size_t)(mrow0 + v) * NSTATES] = c[v] + bias;
}

// ---------------------------------------------------------------------------
// Phase 2: per-row statevector simulation fully in LDS.
// block = 256 threads, one block per row of H (2048 blocks).
// ---------------------------------------------------------------------------
__global__ __launch_bounds__(256) void qml_quantum(
    const float* __restrict__ H,       // (2048,4096) or nullptr -> compute in-block
    const float* __restrict__ x,
    const float* __restrict__ W_in,
    const float* __restrict__ b_in,
    const float* __restrict__ qw,      // (4,12,3)
    const float* __restrict__ W_out,   // (96,12)
    const float* __restrict__ b_out,   // (96,)
    float* __restrict__ out)           // (64,96,32)
{
    __shared__ float2 st[NSTATES];             // 32 KB statevector
    __shared__ float  gates[NLAYERS * NQ][8];  // 48 x (2x2 complex)
    __shared__ float  red[256];
    __shared__ float  zpart[256][NQ];
    __shared__ float  zv[NQ];
    __shared__ float  xrow[LDIM];

    const int tid = threadIdx.x;
    const int row = blockIdx.x;

    // --- precompute Rot(phi,theta,omega) matrices ---
    if (tid < NLAYERS * NQ) {
        const float phi = qw[tid * 3 + 0];
        const float th  = qw[tid * 3 + 1];
        const float om  = qw[tid * 3 + 2];
        float sn, cs, sa, ca, sb, cb;
        sincosf(0.5f * th,        &sn, &cs);
        sincosf(0.5f * (phi + om), &sa, &ca);
        sincosf(0.5f * (phi - om), &sb, &cb);
        gates[tid][0] =  ca * cs;  gates[tid][1] = -sa * cs;  // U00 = e^{-i(phi+om)/2} c
        gates[tid][2] = -cb * sn;  gates[tid][3] = -sb * sn;  // U01 = -e^{ i(phi-om)/2} s
        gates[tid][4] =  cb * sn;  gates[tid][5] = -sb * sn;  // U10 =  e^{-i(phi-om)/2} s
        gates[tid][6] =  ca * cs;  gates[tid][7] =  sa * cs;  // U11 =  e^{ i(phi+om)/2} c
    }

    // --- load/compute h row, accumulate sum of squares ---
    float sq = 0.f;
    if (H) {
        const float* hr = H + (size_t)row * NSTATES;
        for (int i = tid; i < NSTATES; i += 256) {
            const float v = hr[i];
            sq += v * v;
            st[i].x = v; st[i].y = 0.f;
        }
    } else {
        // fallback: compute h in-block (no workspace)
        if (tid < LDIM)
            xrow[tid] = x[(size_t)(row >> 5) * (LDIM * 32) + tid * 32 + (row & 31)];
        __syncthreads();
        for (int i = tid; i < NSTATES; i += 256) {
            const float* wr = W_in + (size_t)i * LDIM;
            float v = b_in[i] + 1e-6f;
            for (int l = 0; l < LDIM; ++l) v += xrow[l] * wr[l];
            sq += v * v;
            st[i].x = v; st[i].y = 0.f;
        }
    }

    // --- block reduce sum of squares, normalize ---
    red[tid] = sq;
    __syncthreads();
    for (int s2 = 128; s2 > 0; s2 >>= 1) {
        if (tid < s2) red[tid] += red[tid + s2];
        __syncthreads();
    }
    const float inv = 1.0f / sqrtf(red[0]);
    for (int i = tid; i < NSTATES; i += 256) st[i].x *= inv;

    // --- StronglyEntanglingLayers ---
    for (int l = 0; l < NLAYERS; ++l) {
        // Rot on each wire
        for (int w = 0; w < NQ; ++w) {
            __syncthreads();
            const float* g = gates[l * NQ + w];
            const float u00r = g[0], u00i = g[1], u01r = g[2], u01i = g[3];
            const float u10r = g[4], u10i = g[5], u11r = g[6], u11i = g[7];
            const int bpos = NQ - 1 - w;
            const int half = 1 << bpos;
            for (int p = tid; p < NSTATES / 2; p += 256) {
                const int lowm = p & (half - 1);
                const int i0 = ((p - lowm) << 1) | lowm;
                const int i1 = i0 | half;
                const float2 a0 = st[i0], a1 = st[i1];
                float2 n0, n1;
                n0.x = u00r * a0.x - u00i * a0.y + u01r * a1.x - u01i * a1.y;
                n0.y = u00r * a0.y + u00i * a0.x + u01r * a1.y + u01i * a1.x;
                n1.x = u10r * a0.x - u10i * a0.y + u11r * a1.x - u11i * a1.y;
                n1.y = u10r * a0.y + u10i * a0.x + u11r * a1.y + u11i * a1.x;
                st[i0] = n0; st[i1] = n1;
            }
        }
        // CNOT ring with range r
        const int rr = l % (NQ - 1) + 1;
        for (int w = 0; w < NQ; ++w) {
            __syncthreads();
            const int cbit = NQ - 1 - w;
            const int tbit = NQ - 1 - ((w + rr) % NQ);
            const int hi = cbit > tbit ? cbit : tbit;
            const int lo = cbit > tbit ? tbit : cbit;
            for (int p = tid; p < NSTATES / 4; p += 256) {
                const int lml = p & ((1 << lo) - 1);
                const int t1  = ((p - lml) << 1) | lml;
                const int lmh = t1 & ((1 << hi) - 1);
                const int i00 = ((t1 - lmh) << 1) | lmh;
                const int ia = i00 | (1 << cbit);            // c=1, t=0
                const int ib = ia  | (1 << tbit);            // c=1, t=1
                const float2 tv = st[ia]; st[ia] = st[ib]; st[ib] = tv;
            }
        }
    }
    __syncthreads();

    // --- PauliZ expectations ---
    float acc[NQ];
#pragma unroll
    for (int q = 0; q < NQ; ++q) acc[q] = 0.f;
    for (int i = tid; i < NSTATES; i += 256) {
        const float2 a = st[i];
        const float pr = a.x * a.x + a.y * a.y;
#pragma unroll
        for (int q = 0; q < NQ; ++q)
            acc[q] += (i & (1 << (NQ - 1 - q))) ? -pr : pr;
    }
#pragma unroll
    for (int q = 0; q < NQ; ++q) zpart[tid][q] = acc[q];
    __syncthreads();
    if (tid < NQ) {
        float s = 0.f;
        for (int t = 0; t < 256; ++t) s += zpart[t][tid];
        zv[tid] = s;
    }
    __syncthreads();

    // --- output projection, written transposed to (64,96,32) ---
    if (tid < FOUT) {
        float s = b_out[tid];
#pragma unroll
        for (int q = 0; q < NQ; ++q) s += zv[q] * W_out[tid * NQ + q];
        const int bi = row >> 5, mi = row & 31;
        out[(size_t)bi * (FOUT * 32) + tid * 32 + mi] = s;
    }
}

// ---------------------------------------------------------------------------
extern "C" void kernel_launch(void* const* d_in, const int* in_sizes, int n_in,
                              void* d_out, int out_size, void* d_ws, size_t ws_size,
                              hipStream_t stream) {
    const float* x     = (const float*)d_in[0];
    const float* W_in  = (const float*)d_in[1];
    const float* b_in  = (const float*)d_in[2];
    const float* qw    = (const float*)d_in[3];
    const float* W_out = (const float*)d_in[4];
    const float* b_out = (const float*)d_in[5];
    float* out = (float*)d_out;

    const size_t needH = (size_t)BM * NSTATES * sizeof(float);   // 32 MB
    float* H = (ws_size >= needH) ? (float*)d_ws : nullptr;

    if (H) {
        // 32768 tiles, 8 waves (one tile each) per 256-thread block
        qml_gemm_wmma<<<4096, 256, 0, stream>>>(x, W_in, b_in, H);
    }
    qml_quantum<<<BM, 256, 0, stream>>>(H, x, W_in, b_in, qw, W_out, b_out, out);
}